// Band_SpectralMix_12790412607959
// MI455X (gfx1250) — compile-verified
//
#include <hip/hip_runtime.h>
#include <math.h>

typedef __attribute__((ext_vector_type(2))) float v2f;
typedef __attribute__((ext_vector_type(8))) float v8f;

#define BB 64
#define TT 512
#define CC 1024
#define MTOK (BB * TT)
#define KC 16          // K-chunk
#define PAD 20         // LDS row stride (floats): 80B = 16B-aligned, gcd(20,64)=4 -> conflict-free

// Async global->LDS copy of 16 bytes (gfx1250 ASYNC path, tracked by ASYNCcnt).
__device__ __forceinline__ void async_copy_b128(const float* gptr, const float* lptr) {
    unsigned lds_off = (unsigned)(uintptr_t)lptr;          // low 32 bits of flat = LDS offset
    unsigned long long ga = (unsigned long long)(uintptr_t)gptr;
    asm volatile("global_load_async_to_lds_b128 %0, %1, off"
                 :: "v"(lds_off), "v"(ga) : "memory");
}
__device__ __forceinline__ void wait_async() {
    asm volatile("s_wait_asynccnt 0" ::: "memory");
}

// ---------------------------------------------------------------------------
// Fused k/v/r projection:
//   xx = x[t-1] (0 at t==0), d = x - xx
//   k  = (xx + d*mix_k) @ Wk^T      -> kbuf
//   v  = (xx + d*mix_v) @ Wv^T      -> vbuf
//   sr = sigmoid((xx + d*mix_r) @ Wr^T) -> srbuf
// Block 256 thr (8 waves), tile 128(M) x 64(N), K-chunk 16.
// Grid: x = N-blocks (fastest -> 16 blocks share one A stripe via L2), y = M.
// ---------------------------------------------------------------------------
__global__ __launch_bounds__(256)
void rwkv_kvr_kernel(const float* __restrict__ x,
                     const float* __restrict__ Wk, const float* __restrict__ Wv,
                     const float* __restrict__ Wr,
                     const float* __restrict__ mixk, const float* __restrict__ mixv,
                     const float* __restrict__ mixr,
                     float* __restrict__ kbuf, float* __restrict__ vbuf,
                     float* __restrict__ srbuf) {
    __shared__ float Sxx[128 * PAD];
    __shared__ float Sd [128 * PAD];
    __shared__ float Swk[64 * PAD];
    __shared__ float Swv[64 * PAD];
    __shared__ float Swr[64 * PAD];
    __shared__ float SmK[KC], SmV[KC], SmR[KC];

    const int tid   = threadIdx.x;
    const int lane  = tid & 31;
    const int wave  = tid >> 5;
    const int waveM = wave >> 1;          // 0..3
    const int waveN = wave & 1;           // 0..1
    const int l16   = lane & 15;
    const int hi    = lane >> 4;

    const int n0 = blockIdx.x * 64;       // fastest dim: output columns
    const int m0 = blockIdx.y * 128;      // token rows

    v8f ak_[2][2], av_[2][2], ar_[2][2];
#pragma unroll
    for (int i = 0; i < 2; ++i)
#pragma unroll
        for (int j = 0; j < 2; ++j)
#pragma unroll
            for (int e = 0; e < 8; ++e) { ak_[i][j][e] = 0.f; av_[i][j][e] = 0.f; ar_[i][j][e] = 0.f; }

    // W-tile copy slots: 64x16 floats = 256 float4 = 1 per thread per matrix
    const int wdl = tid >> 2;             // 0..63
    const int wkq = (tid & 3) * 4;        // 0,4,8,12

    for (int k0 = 0; k0 < CC; k0 += KC) {
        // ---- async-stage the three W tiles (pure copies)
        async_copy_b128(Wk + (size_t)(n0 + wdl) * CC + k0 + wkq, &Swk[wdl * PAD + wkq]);
        async_copy_b128(Wv + (size_t)(n0 + wdl) * CC + k0 + wkq, &Swv[wdl * PAD + wkq]);
        async_copy_b128(Wr + (size_t)(n0 + wdl) * CC + k0 + wkq, &Swr[wdl * PAD + wkq]);

        // ---- stage xx / d tiles (needs arithmetic): 128x16 = 512 float4, 2/thread
#pragma unroll
        for (int i = 0; i < 2; ++i) {
            int idx = i * 256 + tid;
            int ml = idx >> 2, kq = (idx & 3) * 4;
            int row = m0 + ml;
            int t = row & (TT - 1);
            const float4 xc = *(const float4*)(x + (size_t)row * CC + k0 + kq);
            float4 xp = make_float4(0.f, 0.f, 0.f, 0.f);
            if (t > 0) xp = *(const float4*)(x + (size_t)(row - 1) * CC + k0 + kq);
            float4 d4 = make_float4(xc.x - xp.x, xc.y - xp.y, xc.z - xp.z, xc.w - xp.w);
            *(float4*)(&Sxx[ml * PAD + kq]) = xp;
            *(float4*)(&Sd [ml * PAD + kq]) = d4;
        }
        // ---- stage mix scalars for this K-chunk
        if (tid < KC) {
            SmK[tid] = mixk[k0 + tid];
            SmV[tid] = mixv[k0 + tid];
            SmR[tid] = mixr[k0 + tid];
        }
        wait_async();
        __syncthreads();

        // ---- 4 K-substeps of 4: 2tm x 3mats x 2tn = 12 wmma each -> 48/chunk
#pragma unroll
        for (int kk = 0; kk < 4; ++kk) {
            const int kb = kk * 4 + 2 * hi;
            const float mk0 = SmK[kb], mk1 = SmK[kb + 1];
            const float mv0 = SmV[kb], mv1 = SmV[kb + 1];
            const float mr0 = SmR[kb], mr1 = SmR[kb + 1];

            v2f bk[2], bv[2], br[2];
#pragma unroll
            for (int tn = 0; tn < 2; ++tn) {
                const int n = waveN * 32 + tn * 16 + l16;
                bk[tn].x = Swk[n * PAD + kb]; bk[tn].y = Swk[n * PAD + kb + 1];
                bv[tn].x = Swv[n * PAD + kb]; bv[tn].y = Swv[n * PAD + kb + 1];
                br[tn].x = Swr[n * PAD + kb]; br[tn].y = Swr[n * PAD + kb + 1];
            }
#pragma unroll
            for (int tm = 0; tm < 2; ++tm) {
                const int m = waveM * 32 + tm * 16 + l16;
                const float xx0 = Sxx[m * PAD + kb], xx1 = Sxx[m * PAD + kb + 1];
                const float d0  = Sd [m * PAD + kb], d1  = Sd [m * PAD + kb + 1];
                v2f ak; ak.x = fmaf(d0, mk0, xx0); ak.y = fmaf(d1, mk1, xx1);
                v2f av; av.x = fmaf(d0, mv0, xx0); av.y = fmaf(d1, mv1, xx1);
                v2f ar; ar.x = fmaf(d0, mr0, xx0); ar.y = fmaf(d1, mr1, xx1);
#pragma unroll
                for (int tn = 0; tn < 2; ++tn) {
                    ak_[tm][tn] = __builtin_amdgcn_wmma_f32_16x16x4_f32(
                        false, ak, false, bk[tn], (short)0, ak_[tm][tn], false, false);
                    av_[tm][tn] = __builtin_amdgcn_wmma_f32_16x16x4_f32(
                        false, av, false, bv[tn], (short)0, av_[tm][tn], false, false);
                    ar_[tm][tn] = __builtin_amdgcn_wmma_f32_16x16x4_f32(
                        false, ar, false, br[tn], (short)0, ar_[tm][tn], false, false);
                }
            }
        }
        __syncthreads();
    }

    // ---- epilogue
#pragma unroll
    for (int tm = 0; tm < 2; ++tm) {
#pragma unroll
        for (int tn = 0; tn < 2; ++tn) {
            const int n = n0 + waveN * 32 + tn * 16 + l16;
#pragma unroll
            for (int i = 0; i < 8; ++i) {
                const size_t m = m0 + waveM * 32 + tm * 16 + hi * 8 + i;
                kbuf[m * CC + n] = ak_[tm][tn][i];
                vbuf[m * CC + n] = av_[tm][tn][i];
                float r = ar_[tm][tn][i];
                srbuf[m * CC + n] = 1.0f / (1.0f + __expf(-r));
            }
        }
    }
}

// ---------------------------------------------------------------------------
// Plain GEMM: out[n,d] = sum_c A[n,c] * W[d,c]   (output projection)
// ---------------------------------------------------------------------------
__global__ __launch_bounds__(256)
void rwkv_gemm_kernel(const float* __restrict__ A, const float* __restrict__ W,
                      float* __restrict__ out) {
    __shared__ float As[128 * PAD];
    __shared__ float Ws[64 * PAD];

    const int tid   = threadIdx.x;
    const int lane  = tid & 31;
    const int wave  = tid >> 5;
    const int waveM = wave >> 1;
    const int waveN = wave & 1;
    const int l16   = lane & 15;
    const int hi    = lane >> 4;

    const int n0 = blockIdx.x * 64;
    const int m0 = blockIdx.y * 128;

    v8f acc[2][2];
#pragma unroll
    for (int i = 0; i < 2; ++i)
#pragma unroll
        for (int j = 0; j < 2; ++j)
#pragma unroll
            for (int e = 0; e < 8; ++e) acc[i][j][e] = 0.0f;

    const int wdl = tid >> 2;
    const int wkq = (tid & 3) * 4;

    for (int k0 = 0; k0 < CC; k0 += KC) {
        async_copy_b128(W + (size_t)(n0 + wdl) * CC + k0 + wkq, &Ws[wdl * PAD + wkq]);
#pragma unroll
        for (int i = 0; i < 2; ++i) {
            int idx = i * 256 + tid;
            int ml = idx >> 2, kq = (idx & 3) * 4;
            const float4 a4 = *(const float4*)(A + (size_t)(m0 + ml) * CC + k0 + kq);
            *(float4*)(&As[ml * PAD + kq]) = a4;
        }
        wait_async();
        __syncthreads();

#pragma unroll
        for (int kk = 0; kk < 4; ++kk) {
            const int kb = kk * 4 + 2 * hi;
#pragma unroll
            for (int tm = 0; tm < 2; ++tm) {
                const int m = waveM * 32 + tm * 16 + l16;
                v2f a; a.x = As[m * PAD + kb]; a.y = As[m * PAD + kb + 1];
#pragma unroll
                for (int tn = 0; tn < 2; ++tn) {
                    const int n = waveN * 32 + tn * 16 + l16;
                    v2f b; b.x = Ws[n * PAD + kb]; b.y = Ws[n * PAD + kb + 1];
                    acc[tm][tn] = __builtin_amdgcn_wmma_f32_16x16x4_f32(
                        false, a, false, b, (short)0, acc[tm][tn], false, false);
                }
            }
        }
        __syncthreads();
    }

#pragma unroll
    for (int tm = 0; tm < 2; ++tm)
#pragma unroll
        for (int tn = 0; tn < 2; ++tn) {
            const int n = n0 + waveN * 32 + tn * 16 + l16;
#pragma unroll
            for (int i = 0; i < 8; ++i) {
                const size_t m = m0 + waveM * 32 + tm * 16 + hi * 8 + i;
                out[m * CC + n] = acc[tm][tn][i];
            }
        }
}

// ---------------------------------------------------------------------------
// WKV scan: one thread per (b, c); stable num/den recurrence; y over k in place.
// ---------------------------------------------------------------------------
__global__ __launch_bounds__(256)
void rwkv_wkv_kernel(const float* __restrict__ time_decay,
                     const float* __restrict__ time_first,
                     float* __restrict__ kbuf, const float* __restrict__ vbuf) {
    const int idx = blockIdx.x * blockDim.x + threadIdx.x;
    const int b = idx / CC;
    const int c = idx % CC;
    const float w = -__expf(time_decay[c]);
    const float u = time_first[c];

    float aa = 0.0f, bb = 0.0f, pp = -1e38f;
    size_t base = (size_t)b * TT * CC + c;
    for (int t = 0; t < TT; ++t) {
        size_t off = base + (size_t)t * CC;
        float kt = kbuf[off];
        float vt = vbuf[off];

        float ww = u + kt;
        float q  = fmaxf(pp, ww);
        float e1 = __expf(pp - q);
        float e2 = __expf(ww - q);
        float y  = (e1 * aa + e2 * vt) / (e1 * bb + e2);

        float ww2 = pp + w;
        float q2  = fmaxf(ww2, kt);
        float f1  = __expf(ww2 - q2);
        float f2  = __expf(kt - q2);
        aa = f1 * aa + f2 * vt;
        bb = f1 * bb + f2;
        pp = q2;

        kbuf[off] = y;
    }
}

// gated = sr * y  (float4)
__global__ __launch_bounds__(256)
void rwkv_gate_kernel(const float* __restrict__ sr, const float* __restrict__ y,
                      float* __restrict__ gated) {
    const size_t i = ((size_t)blockIdx.x * blockDim.x + threadIdx.x) * 4;
    const float4 s = *(const float4*)(sr + i);
    const float4 v = *(const float4*)(y + i);
    float4 g;
    g.x = s.x * v.x; g.y = s.y * v.y; g.z = s.z * v.z; g.w = s.w * v.w;
    *(float4*)(gated + i) = g;
}

extern "C" void kernel_launch(void* const* d_in, const int* in_sizes, int n_in,
                              void* d_out, int out_size, void* d_ws, size_t ws_size,
                              hipStream_t stream) {
    const float* x          = (const float*)d_in[0];
    const float* time_decay = (const float*)d_in[1];
    const float* time_first = (const float*)d_in[2];
    const float* time_mix_k = (const float*)d_in[3];
    const float* time_mix_v = (const float*)d_in[4];
    const float* time_mix_r = (const float*)d_in[5];
    const float* Wk         = (const float*)d_in[6];
    const float* Wv         = (const float*)d_in[7];
    const float* Wr         = (const float*)d_in[8];
    const float* Wo         = (const float*)d_in[9];
    float* out = (float*)d_out;

    const size_t elems = (size_t)MTOK * CC;
    float* kbuf = (float*)d_ws;          // k -> y
    float* vbuf = kbuf + elems;          // v -> gated

    dim3 blk(256);
    dim3 grd(CC / 64, MTOK / 128);       // N fastest: 16 x 256

    // fused k / v / sigmoid(r); sr parked in d_out
    rwkv_kvr_kernel<<<grd, blk, 0, stream>>>(x, Wk, Wv, Wr,
                                             time_mix_k, time_mix_v, time_mix_r,
                                             kbuf, vbuf, out);
    // WKV recurrence (y over kbuf)
    rwkv_wkv_kernel<<<(BB * CC) / 256, 256, 0, stream>>>(time_decay, time_first, kbuf, vbuf);
    // gated = sr * y  (over vbuf)
    rwkv_gate_kernel<<<(unsigned)(elems / 4 / 256), 256, 0, stream>>>(out, kbuf, vbuf);
    // out = gated @ Wo^T
    rwkv_gemm_kernel<<<grd, blk, 0, stream>>>(vbuf, Wo, out);
}